// GINConvModule_74861279969841
// MI455X (gfx1250) — compile-verified
//
#include <hip/hip_runtime.h>

typedef __attribute__((ext_vector_type(2))) float v2f;
typedef __attribute__((ext_vector_type(8))) float v8f;

#define D 128

// ---------------------------------------------------------------------------
// Kernel 1: zero the aggregation buffer (float4 stores).
// ---------------------------------------------------------------------------
__global__ __launch_bounds__(256) void gin_zero(float* __restrict__ agg, int n4) {
  int i = blockIdx.x * blockDim.x + threadIdx.x;
  if (i < n4) ((float4*)agg)[i] = make_float4(0.f, 0.f, 0.f, 0.f);
}

// ---------------------------------------------------------------------------
// Kernel 2: edge scatter-add. One wave32 per edge; lane l covers features
// [4l, 4l+4). Gather of x[src] is one coalesced 512B burst per edge (L2-hit:
// x is 51MB, L2 is 192MB). Scatter via f32 atomics into agg (L2-resident).
// ---------------------------------------------------------------------------
__global__ __launch_bounds__(256) void gin_scatter(const float* __restrict__ x,
                                                   const long long* __restrict__ ei,
                                                   float* __restrict__ agg,
                                                   int n_edges) {
  int e    = (int)((blockIdx.x * (unsigned)blockDim.x + threadIdx.x) >> 5);
  int lane = threadIdx.x & 31;
  if (e >= n_edges) return;
  long long src = ei[e];                       // edge_index[0][e]
  long long dst = ei[(size_t)n_edges + e];     // edge_index[1][e]
  float4 v = ((const float4*)(x + (size_t)src * D))[lane];
  float* ad = agg + (size_t)dst * D + lane * 4;
  atomicAdd(ad + 0, v.x);
  atomicAdd(ad + 1, v.y);
  atomicAdd(ad + 2, v.z);
  atomicAdd(ad + 3, v.w);
}

// ---------------------------------------------------------------------------
// Kernel 3: fused MLP  out = relu((x+agg)@W1 + b1) @ W2 + b2
// 128 threads = 4 waves; wave w computes rows [m0, m0+16) for all 128 cols
// using V_WMMA_F32_16X16X4_F32 (exact f32). Layer-1 output staged in LDS
// (row stride 132 floats -> conflict-free transposed A reads for layer 2).
// ---------------------------------------------------------------------------
__global__ __launch_bounds__(128) void gin_mlp(const float* __restrict__ x,
                                               const float* __restrict__ agg,
                                               const float* __restrict__ W1,
                                               const float* __restrict__ b1,
                                               const float* __restrict__ W2,
                                               const float* __restrict__ b2,
                                               float* __restrict__ out,
                                               int n_nodes) {
  __shared__ float hbuf[4][16][D + 4];

  const int w    = threadIdx.x >> 5;
  const int lane = threadIdx.x & 31;
  const int r    = lane & 15;   // matrix row (A) or col (B/C)
  const int kh   = lane >> 4;   // K-half select within the 16x4 tile
  const int m0   = blockIdx.x * 64 + w * 16;

  // Clamp A-row loads; stores are masked at the end instead.
  const int row  = m0 + r;
  const int rowc = row < n_nodes ? row : (n_nodes - 1);
  const float* xr = x   + (size_t)rowc * D;
  const float* ar = agg + (size_t)rowc * D;

  // ---- Layer 1: (x + agg) @ W1 ----
  v8f acc[8] = {};
  for (int k = 0; k < D; k += 4) {
    const int kk = k + 2 * kh;
    v2f a;
    a.x = xr[kk] + ar[kk];
    a.y = xr[kk + 1] + ar[kk + 1];
#pragma unroll
    for (int n = 0; n < 8; ++n) {
      const int col = n * 16 + r;
      v2f b;
      b.x = W1[(size_t)kk * D + col];
      b.y = W1[(size_t)(kk + 1) * D + col];
      acc[n] = __builtin_amdgcn_wmma_f32_16x16x4_f32(
          /*neg_a=*/false, a, /*neg_b=*/false, b,
          /*c_mod=*/(short)0, acc[n], /*reuse_a=*/false, /*reuse_b=*/false);
    }
  }

  // bias + ReLU, stage h in LDS (C layout: row = v + 8*kh, col = n*16 + r)
#pragma unroll
  for (int n = 0; n < 8; ++n) {
    const int col  = n * 16 + r;
    const float bv = b1[col];
#pragma unroll
    for (int v = 0; v < 8; ++v) {
      float hv = acc[n][v] + bv;
      hbuf[w][v + 8 * kh][col] = hv > 0.f ? hv : 0.f;
    }
  }
  __syncthreads();

  // ---- Layer 2: h @ W2 ----
  v8f acc2[8] = {};
  for (int k = 0; k < D; k += 4) {
    const int kk = k + 2 * kh;
    v2f a;
    a.x = hbuf[w][r][kk];
    a.y = hbuf[w][r][kk + 1];
#pragma unroll
    for (int n = 0; n < 8; ++n) {
      const int col = n * 16 + r;
      v2f b;
      b.x = W2[(size_t)kk * D + col];
      b.y = W2[(size_t)(kk + 1) * D + col];
      acc2[n] = __builtin_amdgcn_wmma_f32_16x16x4_f32(
          false, a, false, b, (short)0, acc2[n], false, false);
    }
  }

  // bias + masked store
#pragma unroll
  for (int n = 0; n < 8; ++n) {
    const int col  = n * 16 + r;
    const float bv = b2[col];
#pragma unroll
    for (int v = 0; v < 8; ++v) {
      const int orow = m0 + v + 8 * kh;
      if (orow < n_nodes) out[(size_t)orow * D + col] = acc2[n][v] + bv;
    }
  }
}

// ---------------------------------------------------------------------------
extern "C" void kernel_launch(void* const* d_in, const int* in_sizes, int n_in,
                              void* d_out, int out_size, void* d_ws, size_t ws_size,
                              hipStream_t stream) {
  const float*     x  = (const float*)d_in[0];
  const long long* ei = (const long long*)d_in[1];  // int64 edge_index (2, E)
  const float*     W1 = (const float*)d_in[2];
  const float*     b1 = (const float*)d_in[3];
  const float*     W2 = (const float*)d_in[4];
  const float*     b2 = (const float*)d_in[5];
  float* out = (float*)d_out;

  const int n_nodes = in_sizes[0] / D;  // 100000
  const int n_edges = in_sizes[1] / 2;  // 3200000

  float* agg = (float*)d_ws;            // n_nodes*D floats = 51.2 MB scratch

  const int n4 = n_nodes * (D / 4);
  gin_zero<<<(n4 + 255) / 256, 256, 0, stream>>>(agg, n4);

  // one wave per edge, 8 waves per 256-thread block
  gin_scatter<<<(n_edges + 7) / 8, 256, 0, stream>>>(x, ei, agg, n_edges);

  // 64 rows per 128-thread block (4 waves x 16 rows)
  gin_mlp<<<(n_nodes + 63) / 64, 128, 0, stream>>>(x, agg, W1, b1, W2, b2, out, n_nodes);
}